// VectorQuantizer_80350248173952
// MI455X (gfx1250) — compile-verified
//
#include <hip/hip_runtime.h>
#include <math.h>

// ---------------- problem constants ----------------
#define NROWS   16384      // B*H*W = 16*32*32
#define NGROUPS 4
#define EDIM    64
#define NE_G    4096       // codes per group
#define NTILES  256        // NE_G / 16

// d_out float offsets (outputs concatenated in return order)
#define QUANT_OFF   0                         // 16*256*32*32 = 4194304
#define LOSS_OFF    4194304                   // vq(4)+commit(4)+usage(4) = 12 zeros
#define PERP_OFF    4194316                   // 1
#define MINENC_OFF  4194317                   // 16384*4096 = 67108864
#define IDX_OFF     71303181                  // 16384*4

typedef float v2f __attribute__((ext_vector_type(2)));
typedef float v8f __attribute__((ext_vector_type(8)));

// ---------------------------------------------------------------------------
// Prep: l2-normalize embedding -> embN, pack B into per-lane WMMA layout,
// zero the g=3 index histogram. One wave per codebook row.
// B layout target (f32 4x16 B of V_WMMA_F32_16X16X4_F32):
//   lane L reads float2 slot t: elements e = 4t + 2*(L>>4) + {0,1} of code (L&15)
// ---------------------------------------------------------------------------
__global__ void vq_prep_kernel(const float* __restrict__ emb,
                               float* __restrict__ embN,
                               float* __restrict__ packedB,
                               int* __restrict__ hist)
{
    int gtid = blockIdx.x * blockDim.x + threadIdx.x;
    if (gtid < NE_G) hist[gtid] = 0;

    int row  = gtid >> 5;          // one wave (32 lanes) per codebook row
    int lane = threadIdx.x & 31;
    if (row >= NROWS) return;      // NROWS == total codes (16384)

    const float* er = emb + (size_t)row * EDIM;
    float x0 = er[lane];
    float x1 = er[lane + 32];
    float ss = x0 * x0 + x1 * x1;
    #pragma unroll
    for (int mask = 16; mask >= 1; mask >>= 1)
        ss += __shfl_xor(ss, mask, 32);
    float scale = 1.0f / fmaxf(sqrtf(ss), 1e-12f);
    x0 *= scale; x1 *= scale;

    embN[(size_t)row * EDIM + lane]      = x0;
    embN[(size_t)row * EDIM + lane + 32] = x1;

    int g  = row >> 12;            // group
    int kg = row & (NE_G - 1);     // code within group
    int kt = kg >> 4;              // code tile
    int kc = kg & 15;              // column within tile

    // element e of this code -> packed position
    #pragma unroll
    for (int half = 0; half < 2; ++half) {
        int   e   = lane + 32 * half;
        float val = half ? x1 : x0;
        int t   = e >> 2;
        int r   = e & 3;
        int Lhi = r >> 1;          // which 16-lane half reads it
        int x   = r & 1;           // float2 component
        size_t off = ((((size_t)(g * NTILES + kt)) * 32 + (Lhi * 16 + kc)) * 16 + t) * 2 + x;
        packedB[off] = val;
    }
}

// ---------------------------------------------------------------------------
// Main: per wave, 16 z-rows x 1 group, argmax over 4096 codes via
// chained V_WMMA_F32_16X16X4_F32 (16 steps cover K=64).
// Two code tiles per iteration -> two independent WMMA accumulator chains
// sharing the A operands, doubling in-flight loads and halving the effective
// XDL dependency depth.
// ---------------------------------------------------------------------------
__global__ void vq_argmax_kernel(const float* __restrict__ z,
                                 const float* __restrict__ packedB,
                                 int* __restrict__ idxArr,
                                 float* __restrict__ outIdxF,
                                 int* __restrict__ hist)
{
    const int lane = threadIdx.x & 31;
    const int wid  = blockIdx.x * (blockDim.x >> 5) + (threadIdx.x >> 5);
    const int g    = wid & 3;          // group
    const int rt   = wid >> 2;         // row tile, 0..1023
    const int m    = lane & 15;
    const int hi   = lane >> 4;

    // ---- load + normalize A tile (rows rt*16 .. rt*16+15) ----
    // z_groups element (b, g*64+e, h, w); n = (b*32+h)*32+w
    const int n = rt * 16 + m;
    const int w = n & 31, h = (n >> 5) & 31, b = n >> 10;
    const float* zb = z + (size_t)b * 262144 + (size_t)g * 65536 + h * 32 + w;

    float a[32];
    float ss = 0.f;
    #pragma unroll
    for (int t = 0; t < 16; ++t) {
        int e0 = 4 * t + 2 * hi;
        float x0 = zb[(size_t)e0 * 1024];
        float x1 = zb[(size_t)(e0 + 1) * 1024];
        a[2 * t]     = x0;
        a[2 * t + 1] = x1;
        ss += x0 * x0 + x1 * x1;
    }
    ss += __shfl_xor(ss, 16, 32);      // lanes L and L+16 together hold all 64 elems
    float scale = 1.0f / fmaxf(sqrtf(ss), 1e-12f);
    #pragma unroll
    for (int i = 0; i < 32; ++i) a[i] *= scale;

    float best[8];
    int   bidx[8];
    #pragma unroll
    for (int v = 0; v < 8; ++v) { best[v] = -3.4e38f; bidx[v] = 0; }

    const v2f* bp_grp = (const v2f*)packedB + (size_t)g * NTILES * 512;

    for (int kt = 0; kt < NTILES; kt += 2) {
        const v2f* bp0 = bp_grp + (size_t)kt * 512 + lane * 16;
        const v2f* bp1 = bp0 + 512;
        v8f acc0 = {0.f, 0.f, 0.f, 0.f, 0.f, 0.f, 0.f, 0.f};
        v8f acc1 = {0.f, 0.f, 0.f, 0.f, 0.f, 0.f, 0.f, 0.f};
        #pragma unroll
        for (int t = 0; t < 16; ++t) {
            v2f av; av.x = a[2 * t]; av.y = a[2 * t + 1];
            v2f bv0 = bp0[t];
            v2f bv1 = bp1[t];
            acc0 = __builtin_amdgcn_wmma_f32_16x16x4_f32(
                false, av, false, bv0, (short)0, acc0, false, false);
            acc1 = __builtin_amdgcn_wmma_f32_16x16x4_f32(
                false, av, false, bv1, (short)0, acc1, false, false);
        }
        // D layout: lane holds code column m; VGPR v = row (v + 8*hi).
        // Visit tile kt before kt+1 so strict > keeps the first max index.
        int kbase0 = kt * 16 + m;
        int kbase1 = kbase0 + 16;
        #pragma unroll
        for (int v = 0; v < 8; ++v) {
            float s0 = acc0[v];
            if (s0 > best[v]) { best[v] = s0; bidx[v] = kbase0; }
        }
        #pragma unroll
        for (int v = 0; v < 8; ++v) {
            float s1 = acc1[v];
            if (s1 > best[v]) { best[v] = s1; bidx[v] = kbase1; }
        }
    }

    // cross-lane argmax within each 16-lane half (first-index tie-break)
    #pragma unroll
    for (int v = 0; v < 8; ++v) {
        #pragma unroll
        for (int mask = 8; mask >= 1; mask >>= 1) {
            float ov = __shfl_xor(best[v], mask, 32);
            int   oi = __shfl_xor(bidx[v], mask, 32);
            if (ov > best[v] || (ov == best[v] && oi < bidx[v])) {
                best[v] = ov; bidx[v] = oi;
            }
        }
    }

    if (m == 0) {
        #pragma unroll
        for (int v = 0; v < 8; ++v) {
            int row = rt * 16 + v + 8 * hi;
            int k   = bidx[v];
            idxArr[row * NGROUPS + g]  = k;
            outIdxF[row * NGROUPS + g] = (float)k;
            if (g == NGROUPS - 1) atomicAdd(&hist[k], 1);
        }
    }
}

// ---------------------------------------------------------------------------
// quant: scatter selected normalized codes back to (b, 256, h, w); also the
// 12 zero loss outputs.
// ---------------------------------------------------------------------------
__global__ void vq_quant_kernel(const int* __restrict__ idxArr,
                                const float* __restrict__ embN,
                                float* __restrict__ out)
{
    int t = blockIdx.x * 256 + threadIdx.x;        // 0 .. 4194303
    int w  = t & 31;
    int hh = (t >> 5) & 31;
    int c  = (t >> 10) & 255;
    int b  = t >> 18;
    int g  = c >> 6;
    int e  = c & 63;
    int n  = (b * 32 + hh) * 32 + w;
    int k  = idxArr[n * NGROUPS + g];
    out[t] = embN[((size_t)(g * NE_G + k)) * EDIM + e];
    if (t < 12) out[LOSS_OFF + t] = 0.0f;
}

// ---------------------------------------------------------------------------
// one-hot min_encodings for idx[:, 3]: 16384 x 4096 floats.
// Each 256-thread block lies inside one row (256 | 4096).
// ---------------------------------------------------------------------------
__global__ void vq_minenc_kernel(const int* __restrict__ idxArr,
                                 float* __restrict__ out)
{
    __shared__ int s_k;
    size_t t  = (size_t)blockIdx.x * 256 + threadIdx.x;
    int   row = (int)(t >> 12);
    if (threadIdx.x == 0) s_k = idxArr[row * NGROUPS + (NGROUPS - 1)];
    __syncthreads();
    int col = (int)t & (NE_G - 1);
    out[t] = (col == s_k) ? 1.0f : 0.0f;
}

// ---------------------------------------------------------------------------
// perplexity = exp(-sum p*log(p+1e-10)), p = hist/16384
// ---------------------------------------------------------------------------
__global__ void vq_perp_kernel(const int* __restrict__ hist,
                               float* __restrict__ out)
{
    __shared__ float s[256];
    float acc = 0.f;
    for (int bin = threadIdx.x; bin < NE_G; bin += 256) {
        float p = (float)hist[bin] * (1.0f / (float)NROWS);
        acc -= p * logf(p + 1e-10f);
    }
    s[threadIdx.x] = acc;
    __syncthreads();
    for (int st = 128; st > 0; st >>= 1) {
        if (threadIdx.x < st) s[threadIdx.x] += s[threadIdx.x + st];
        __syncthreads();
    }
    if (threadIdx.x == 0) out[0] = expf(s[0]);
}

// ---------------------------------------------------------------------------
extern "C" void kernel_launch(void* const* d_in, const int* in_sizes, int n_in,
                              void* d_out, int out_size, void* d_ws, size_t ws_size,
                              hipStream_t stream)
{
    const float* z   = (const float*)d_in[0];   // (16, 256, 32, 32)
    const float* emb = (const float*)d_in[1];   // (16384, 64)
    float* out = (float*)d_out;

    char* ws = (char*)d_ws;
    float* embN    = (float*)ws;                             // 4 MB
    float* packedB = (float*)(ws + (4u << 20));              // 4 MB
    int*   idxArr  = (int*)(ws + (8u << 20));                // 256 KB
    int*   hist    = (int*)(ws + (8u << 20) + (256u << 10)); // 16 KB

    // normalize codebook + pack B + zero hist (16384 waves)
    vq_prep_kernel<<<2048, 256, 0, stream>>>(emb, embN, packedB, hist);

    // WMMA argmax: 4096 waves (1024 row tiles x 4 groups), 8 waves/block
    vq_argmax_kernel<<<512, 256, 0, stream>>>(z, packedB, idxArr,
                                              out + (size_t)IDX_OFF, hist);

    // quant scatter + zero losses
    vq_quant_kernel<<<16384, 256, 0, stream>>>(idxArr, embN, out);

    // one-hot encodings (67,108,864 elements)
    vq_minenc_kernel<<<262144, 256, 0, stream>>>(idxArr, out + (size_t)MINENC_OFF);

    // perplexity scalar
    vq_perp_kernel<<<1, 256, 0, stream>>>(hist, out + (size_t)PERP_OFF);
}